// ConcatAtten_25769803882
// MI455X (gfx1250) — compile-verified
//
#include <hip/hip_runtime.h>
#include <hip/hip_bf16.h>

// ---------------------------------------------------------------------------
// ConcatAtten (additive / Bahdanau attention), MI455X gfx1250, wave32 + WMMA
//   B=4, TQ=TK=512, E=128, F=2E=256
//   Qp = query @ W1            [B*512,128]   (WMMA f32 16x16x4)
//   Kp = key   @ W2            [B*512,128]   (WMMA f32 16x16x4)
//   S[b,t,q] = sum_e vc[e]*tanh(Kp[b,t,e]+Qp[b,q,e]); softmax over q
//   out[b,d,q] = sum_t value[b,d,t]*attens[b,t,q]     (WMMA f32 16x16x4)
// ---------------------------------------------------------------------------

#define B_   4
#define TQ_  512
#define TK_  512
#define E_   128
#define F_   256   // 2*E
#define TT_  4     // t-rows per score block

typedef float v2f __attribute__((ext_vector_type(2)));
typedef float v8f __attribute__((ext_vector_type(8)));

// ---------------- WMMA f32 16x16x4 operand helpers (ISA 7.12.2) -------------
// A (16x4, MxK): lanes 0-15 -> M=lane, K=k0,k0+1 ; lanes 16-31 -> M=lane-16, K=k0+2,k0+3
__device__ __forceinline__ v2f load_a_16x4(const float* A, int lda, int m0, int k0, int lane) {
    const int r  = m0 + (lane & 15);
    const int kk = k0 + ((lane >> 4) << 1);
    const float* p = A + (size_t)r * lda + kk;
    v2f v; v.x = p[0]; v.y = p[1];
    return v;
}

// B (4x16, KxN): VGPR0 lanes0-15 = row k0, lanes16-31 = row k0+2; VGPR1 = k0+1 / k0+3
__device__ __forceinline__ v2f load_b_4x16(const float* Bm, int ldb, int k0, int n0, int lane) {
    const int n  = n0 + (lane & 15);
    const int kk = k0 + ((lane >> 4) << 1);
    v2f v;
    v.x = Bm[(size_t)kk       * ldb + n];
    v.y = Bm[(size_t)(kk + 1) * ldb + n];
    return v;
}

// C/D (16x16 f32): lanes 0-15 -> rows 0..7 (vgpr i = row i), lanes 16-31 -> rows 8..15
__device__ __forceinline__ void store_d_16x16(float* D, int ldd, int m0, int n0, int lane, v8f acc) {
    const int col   = n0 + (lane & 15);
    const int rbase = m0 + ((lane >> 4) << 3);
#pragma unroll
    for (int i = 0; i < 8; ++i)
        D[(size_t)(rbase + i) * ldd + col] = acc[i];
}

__device__ __forceinline__ v8f wmma_f32_4(v2f a, v2f b, v8f c) {
    return __builtin_amdgcn_wmma_f32_16x16x4_f32(
        /*neg_a=*/false, a, /*neg_b=*/false, b,
        /*c_mod=*/(short)0, c, /*reuse_a=*/false, /*reuse_b=*/false);
}

// ---------------- fast device math -----------------------------------------
__device__ __forceinline__ float fast_rcp(float x) {
#if __has_builtin(__builtin_amdgcn_rcpf)
    return __builtin_amdgcn_rcpf(x);      // v_rcp_f32
#else
    return 1.0f / x;
#endif
}

__device__ __forceinline__ float fast_exp(float x) {
#if __has_builtin(__builtin_amdgcn_exp2f)
    return __builtin_amdgcn_exp2f(x * 1.4426950408889634f);   // v_exp_f32
#else
    return __expf(x);
#endif
}

__device__ __forceinline__ float fast_tanh(float x) {
#if __has_builtin(__builtin_amdgcn_tanhf)
    return __builtin_amdgcn_tanhf(x);     // v_tanh_f32
#elif __has_builtin(__builtin_amdgcn_exp2f)
    // tanh(x) = 1 - 2/(exp2(2x*log2 e)+1); saturates correctly at +/-inf.
    // rcp instead of div: avoids the precise v_div_scale/v_div_fmas sequence.
    float e = __builtin_amdgcn_exp2f(x * 2.885390081777927f);
    return 1.0f - 2.0f * fast_rcp(e + 1.0f);
#else
    return tanhf(x);
#endif
}

// ---------------------------------------------------------------------------
// Kernel 1: fused projection of query->Qp and key->Kp.
// Combined row space: rows [0,2048) = query rows, [2048,4096) = key rows.
// One wave (32 threads) computes a 16(M) x 64(N) strip; K = 256 step 4.
// grid = (4096/16, 128/64) = (256, 2), block = 32.
// ---------------------------------------------------------------------------
__global__ __launch_bounds__(32)
void proj_wmma_kernel(const float* __restrict__ query,
                      const float* __restrict__ key,
                      const float* __restrict__ W1,
                      const float* __restrict__ W2,
                      float* __restrict__ Qp,
                      float* __restrict__ Kp) {
    const int lane = threadIdx.x;
    const int mt   = blockIdx.x;           // combined M-tile
    const int n0   = blockIdx.y * 64;      // N strip base

    int row0 = mt * 16;
    const float* A;
    const float* W;
    float*       D;
    if (row0 < B_ * TQ_) {
        A = query;           W = W1; D = Qp;
    } else {
        row0 -= B_ * TQ_;
        A = key;             W = W2; D = Kp;
    }

    v8f acc0 = {}, acc1 = {}, acc2 = {}, acc3 = {};

    for (int k0 = 0; k0 < F_; k0 += 4) {
        v2f a  = load_a_16x4(A, F_, row0, k0, lane);
        v2f b0 = load_b_4x16(W, E_, k0, n0 +  0, lane);
        v2f b1 = load_b_4x16(W, E_, k0, n0 + 16, lane);
        v2f b2 = load_b_4x16(W, E_, k0, n0 + 32, lane);
        v2f b3 = load_b_4x16(W, E_, k0, n0 + 48, lane);
        acc0 = wmma_f32_4(a, b0, acc0);
        acc1 = wmma_f32_4(a, b1, acc1);
        acc2 = wmma_f32_4(a, b2, acc2);
        acc3 = wmma_f32_4(a, b3, acc3);
    }

    store_d_16x16(D, E_, row0, n0 +  0, lane, acc0);
    store_d_16x16(D, E_, row0, n0 + 16, lane, acc1);
    store_d_16x16(D, E_, row0, n0 + 32, lane, acc2);
    store_d_16x16(D, E_, row0, n0 + 48, lane, acc3);
}

// ---------------------------------------------------------------------------
// Kernel 2: score + tanh + vc projection + softmax over q.
// block = 256 threads handles TT_=4 consecutive t rows of one batch:
//   - 4 K-rows staged in LDS, Q rows read once (float4) and reused 4x
//   - each thread owns q = tid and q = tid+256, 4 accumulators each
//   - per-t LDS-tree softmax over the 512 q values, coalesced store
// grid = (TK/TT, B).
// ---------------------------------------------------------------------------
__global__ __launch_bounds__(256)
void score_softmax_kernel(const float* __restrict__ Qp,
                          const float* __restrict__ Kp,
                          const float* __restrict__ vc,
                          float* __restrict__ attens) {
    const int tid = threadIdx.x;
    const int t0  = blockIdx.x * TT_;
    const int b   = blockIdx.y;

    __shared__ float krow[TT_][E_];
    __shared__ float vcs[E_];
    __shared__ float sv[TT_][TQ_];
    __shared__ float red[256];

    if (tid < E_) vcs[tid] = vc[tid];
    for (int i = tid; i < TT_ * E_; i += 256)
        krow[i >> 7][i & (E_ - 1)] =
            Kp[((size_t)(b * TK_ + t0 + (i >> 7))) * E_ + (i & (E_ - 1))];
    __syncthreads();

#pragma unroll
    for (int qi = 0; qi < 2; ++qi) {
        const int q = tid + qi * 256;
        const float4* qr = (const float4*)(Qp + ((size_t)(b * TQ_ + q)) * E_);
        float acc[TT_];
#pragma unroll
        for (int t = 0; t < TT_; ++t) acc[t] = 0.0f;

        for (int e4 = 0; e4 < E_ / 4; ++e4) {
            const float4 qv = qr[e4];
            const int e = e4 * 4;
            const float v0 = vcs[e + 0], v1 = vcs[e + 1];
            const float v2 = vcs[e + 2], v3 = vcs[e + 3];
#pragma unroll
            for (int t = 0; t < TT_; ++t) {
                acc[t] += v0 * fast_tanh(krow[t][e + 0] + qv.x);
                acc[t] += v1 * fast_tanh(krow[t][e + 1] + qv.y);
                acc[t] += v2 * fast_tanh(krow[t][e + 2] + qv.z);
                acc[t] += v3 * fast_tanh(krow[t][e + 3] + qv.w);
            }
        }
#pragma unroll
        for (int t = 0; t < TT_; ++t) sv[t][q] = acc[t];
    }
    __syncthreads();

    // softmax over q (512 values) for each of the TT_ rows
    for (int t = 0; t < TT_; ++t) {
        red[tid] = fmaxf(sv[t][tid], sv[t][tid + 256]);
        __syncthreads();
#pragma unroll
        for (int s = 128; s > 0; s >>= 1) {
            if (tid < s) red[tid] = fmaxf(red[tid], red[tid + s]);
            __syncthreads();
        }
        const float m = red[0];
        __syncthreads();

        const float p0 = fast_exp(sv[t][tid] - m);
        const float p1 = fast_exp(sv[t][tid + 256] - m);
        red[tid] = p0 + p1;
        __syncthreads();
#pragma unroll
        for (int s = 128; s > 0; s >>= 1) {
            if (tid < s) red[tid] += red[tid + s];
            __syncthreads();
        }
        const float inv = fast_rcp(red[0]);
        __syncthreads();   // red reused next t iteration

        float* arow = attens + ((size_t)(b * TK_ + t0 + t)) * TQ_;
        arow[tid]       = p0 * inv;
        arow[tid + 256] = p1 * inv;
    }
}

// ---------------------------------------------------------------------------
// Kernel 3: out[b] = value[b] (256x512) @ attens[b] (512x512), f32 WMMA.
// One wave computes a 16(M) x 64(N) strip; K = 512 step 4.
// grid = (256/16, 512/64, B) = (16, 8, 4), block = 32.
// ---------------------------------------------------------------------------
__global__ __launch_bounds__(32)
void out_gemm_kernel(const float* __restrict__ value,
                     const float* __restrict__ attens,
                     float* __restrict__ out) {
    const int lane = threadIdx.x;
    const int m0   = blockIdx.x * 16;
    const int n0   = blockIdx.y * 64;
    const int b    = blockIdx.z;

    const float* A  = value  + (size_t)b * F_  * TK_;   // [256 x 512]
    const float* Bm = attens + (size_t)b * TK_ * TQ_;   // [512 x 512]
    float*       D  = out    + (size_t)b * F_  * TQ_;   // [256 x 512]

    v8f acc0 = {}, acc1 = {}, acc2 = {}, acc3 = {};

    for (int k0 = 0; k0 < TK_; k0 += 4) {
        v2f a  = load_a_16x4(A, TK_, m0, k0, lane);
        v2f b0 = load_b_4x16(Bm, TQ_, k0, n0 +  0, lane);
        v2f b1 = load_b_4x16(Bm, TQ_, k0, n0 + 16, lane);
        v2f b2 = load_b_4x16(Bm, TQ_, k0, n0 + 32, lane);
        v2f b3 = load_b_4x16(Bm, TQ_, k0, n0 + 48, lane);
        acc0 = wmma_f32_4(a, b0, acc0);
        acc1 = wmma_f32_4(a, b1, acc1);
        acc2 = wmma_f32_4(a, b2, acc2);
        acc3 = wmma_f32_4(a, b3, acc3);
    }

    store_d_16x16(D, TQ_, m0, n0 +  0, lane, acc0);
    store_d_16x16(D, TQ_, m0, n0 + 16, lane, acc1);
    store_d_16x16(D, TQ_, m0, n0 + 32, lane, acc2);
    store_d_16x16(D, TQ_, m0, n0 + 48, lane, acc3);
}

// ---------------------------------------------------------------------------
extern "C" void kernel_launch(void* const* d_in, const int* in_sizes, int n_in,
                              void* d_out, int out_size, void* d_ws, size_t ws_size,
                              hipStream_t stream) {
    const float* query = (const float*)d_in[0];   // [B,TQ,256]
    const float* key   = (const float*)d_in[1];   // [B,TK,256]
    const float* value = (const float*)d_in[2];   // [B,256,TK]
    const float* W1    = (const float*)d_in[3];   // [256,128]
    const float* W2    = (const float*)d_in[4];   // [256,128]
    const float* vc    = (const float*)d_in[5];   // [128]
    float* out = (float*)d_out;                   // [B,256,TQ]

    // workspace: Qp (1MB) | Kp (1MB) | attens (4MB)
    float* Qp     = (float*)d_ws;
    float* Kp     = Qp + (size_t)B_ * TQ_ * E_;
    float* attens = Kp + (size_t)B_ * TK_ * E_;

    // 1) projections (query and key fused into one combined row space)
    {
        dim3 grid((2 * B_ * TQ_) / 16, E_ / 64, 1);
        proj_wmma_kernel<<<grid, 32, 0, stream>>>(query, key, W1, W2, Qp, Kp);
    }
    // 2) additive score + tanh + vc + softmax over q
    {
        dim3 grid(TK_ / TT_, B_, 1);
        score_softmax_kernel<<<grid, 256, 0, stream>>>(Qp, Kp, vc, attens);
    }
    // 3) out = value @ attens
    {
        dim3 grid(F_ / 16, TQ_ / 64, B_);
        out_gemm_kernel<<<grid, 32, 0, stream>>>(value, attens, out);
    }
}